// VectorQuantize_12352325943959
// MI455X (gfx1250) — compile-verified
//
#include <hip/hip_runtime.h>

#define NE      8192        // N_EMBED
#define ROWS    32768       // 32*16*64
#define NELEM   2097152     // 32*16*64*64
#define DECAYF  0.8f
#define EPSF    1e-5f

typedef __attribute__((ext_vector_type(2))) float v2f;
typedef __attribute__((ext_vector_type(8))) float v8f;

// ---- d_out layout (floats) ----
#define O_QUANT   0                      // 2097152  (also used as 'flat' scratch)
#define O_IND     2097152                // 32768
#define O_LOSS    2129920                // 1
#define O_CLUSTER 2129921                // 8192
#define O_EAVG    2138113                // 524288
#define O_EMBED   2662401                // 524288

// ---- workspace layout (floats) ----
#define W_COUNTS  0                      // 8192 (int)
#define W_LOSS    8192                   // 1
#define W_ESUM    8448                   // 524288
#define ZERO_N    532736                 // contiguous zero region [0, 532736)
#define W_ENORM   532736                 // 8192
#define W_IND     540928                 // 32768 (int)
#define W_SMOOTH  573696                 // 8192

__global__ __launch_bounds__(256) void k_zero(float* ws, int n) {
    int e = blockIdx.x * 256 + threadIdx.x;
    if (e < n) ws[e] = 0.0f;
}

// -0.5 * ||embed[:,n]||^2
__global__ __launch_bounds__(256) void k_enorm(const float* __restrict__ embed,
                                               float* __restrict__ enorm) {
    int n = blockIdx.x * 256 + threadIdx.x;
    if (n >= NE) return;
    float s = 0.f;
    #pragma unroll 8
    for (int d = 0; d < 64; ++d) { float v = embed[d * NE + n]; s += v * v; }
    enorm[n] = -0.5f * s;
}

// flat = input + permuted pos_emb (pre_quantize collapses to identity + add)
__global__ __launch_bounds__(256) void k_prequant(const float* __restrict__ input,
                                                  const float* __restrict__ pos,
                                                  float* __restrict__ flat) {
    int e = blockIdx.x * 256 + threadIdx.x;
    if (e >= NELEM) return;
    int W = e & 63, H = (e >> 6) & 63, c = (e >> 12) & 15;
    int t = ((H >> 2) << 4) + (W >> 2);
    int f = (c << 4) + ((H & 3) << 2) + (W & 3);
    flat[e] = input[e] + pos[t * 256 + f];
}

// Argmax_n ( flat_i . e_n - 0.5*|e_n|^2 ) via V_WMMA_F32_16X16X4_F32.
// 8 waves/block; wave = one 16-row M tile. Two 64x16 B tiles staged in LDS
// per stage (software-pipelined global loads). sched_barrier(0) pins all 32
// ds loads before the 32-WMMA block so DS latency is pipelined, not serialized.
__global__ __launch_bounds__(256) void k_argmax(const float* __restrict__ flat,
                                                const float* __restrict__ embed,
                                                const float* __restrict__ enorm,
                                                int* __restrict__ indI,
                                                float* __restrict__ indF) {
    __shared__ float bt[2 * 64 * 16];          // tile0 @ 0, tile1 @ 1024 (bank-aligned)
    const int tid  = threadIdx.x;
    const int wave = tid >> 5, lane = tid & 31;
    const int L15  = lane & 15, half = lane >> 4;
    const int mbase = (blockIdx.x * 8 + wave) * 16;

    // staging addresses: bt[idx] = embed[k][N0 + t*16 + n], idx = t*1024 + k*16 + n
    int sIdx[8], sOff[8];
    #pragma unroll
    for (int rr = 0; rr < 8; ++rr) {
        int idx = tid + rr * 256;
        int t   = idx >> 10;
        int rem = idx & 1023;
        sIdx[rr] = idx;
        sOff[rr] = (rem >> 4) * NE + t * 16 + (rem & 15);
    }

    // A tile: ISA 16x4 f32 layout -> lane L<16 row L holds K={0,1}+4k, lane L+16 holds K={2,3}+4k
    v2f a[16];
    const float* arow = flat + (mbase + L15) * 64 + 2 * half;
    #pragma unroll
    for (int kk = 0; kk < 16; ++kk)
        a[kk] = *(const v2f*)(arow + 4 * kk);

    float bv[8];
    int   bn[8];
    #pragma unroll
    for (int r = 0; r < 8; ++r) { bv[r] = -3.402823466e38f; bn[r] = 0; }

    // prologue: fetch tile pair 0
    float g[8];
    #pragma unroll
    for (int rr = 0; rr < 8; ++rr) g[rr] = embed[sOff[rr]];

    for (int nt = 0; nt < NE / 32; ++nt) {
        const int N0 = nt * 32;
        __syncthreads();                       // previous tiles fully consumed
        #pragma unroll
        for (int rr = 0; rr < 8; ++rr) bt[sIdx[rr]] = g[rr];
        __syncthreads();

        // branch-free prefetch of next tile pair (wraps on last iteration)
        const int Npf = (N0 + 32) & (NE - 1);
        #pragma unroll
        for (int rr = 0; rr < 8; ++rr) g[rr] = embed[Npf + sOff[rr]];

        float ci0 = enorm[N0 + L15];           // fold -0.5|e|^2 into C
        float ci1 = enorm[N0 + 16 + L15];

        __builtin_amdgcn_sched_barrier(0);     // keep prefetch/enorm issued early

        // preload ALL B operands into distinct registers (pipelined ds loads)
        v2f b0[16], b1[16];
        #pragma unroll
        for (int kk = 0; kk < 16; ++kk) {
            const int rowb = (4 * kk + 2 * half) * 16 + L15;
            b0[kk].x = bt[rowb];
            b0[kk].y = bt[rowb + 16];
            b1[kk].x = bt[1024 + rowb];
            b1[kk].y = bt[1024 + rowb + 16];
        }

        __builtin_amdgcn_sched_barrier(0);     // loads may NOT sink past this point

        v8f c0 = { ci0, ci0, ci0, ci0, ci0, ci0, ci0, ci0 };
        v8f c1 = { ci1, ci1, ci1, ci1, ci1, ci1, ci1, ci1 };
        #pragma unroll
        for (int kk = 0; kk < 16; ++kk) {
            c0 = __builtin_amdgcn_wmma_f32_16x16x4_f32(false, a[kk], false, b0[kk],
                                                       (short)0, c0, false, false);
            c1 = __builtin_amdgcn_wmma_f32_16x16x4_f32(false, a[kk], false, b1[kk],
                                                       (short)0, c1, false, false);
        }

        const int n0 = N0 + L15, n1 = n0 + 16;
        #pragma unroll
        for (int r = 0; r < 8; ++r) {          // first-max wins per lane (n increasing)
            if (c0[r] > bv[r]) { bv[r] = c0[r]; bn[r] = n0; }
            if (c1[r] > bv[r]) { bv[r] = c1[r]; bn[r] = n1; }
        }
    }

    // reduce across the 16 lanes sharing each row (halves stay independent)
    #pragma unroll
    for (int m = 1; m <= 8; m <<= 1) {
        #pragma unroll
        for (int r = 0; r < 8; ++r) {
            float ov = __shfl_xor(bv[r], m, 32);
            int   on = __shfl_xor(bn[r], m, 32);
            if (ov > bv[r] || (ov == bv[r] && on < bn[r])) { bv[r] = ov; bn[r] = on; }
        }
    }
    if (L15 == 0) {                            // lane0 -> rows 0..7, lane16 -> rows 8..15
        int rbase = mbase + half * 8;
        #pragma unroll
        for (int r = 0; r < 8; ++r) {
            indI[rbase + r] = bn[r];
            indF[rbase + r] = (float)bn[r];
        }
    }
}

// Per row: segment-sum flat into embed_sum, gather quantize (overwrite flat),
// loss partial, counts. 4 rows per 256-thread block.
__global__ __launch_bounds__(256) void k_scatter(const float* __restrict__ input,
                                                 const float* __restrict__ embed,
                                                 const int* __restrict__ ind,
                                                 float* __restrict__ qflat,
                                                 float* __restrict__ esum,
                                                 int* __restrict__ counts,
                                                 float* __restrict__ lossAcc) {
    __shared__ float wsum[8];
    int tid = threadIdx.x;
    int row = blockIdx.x * 4 + (tid >> 6);
    int d   = tid & 63;
    int n   = ind[row];
    int e   = row * 64 + d;
    float fv = qflat[e];                       // read flat BEFORE overwrite
    atomicAdd(&esum[d * NE + n], fv);
    float qv = embed[d * NE + n];
    qflat[e] = qv;                             // quantize output
    float diff = qv - input[e];
    float v = diff * diff;
    #pragma unroll
    for (int m = 16; m >= 1; m >>= 1) v += __shfl_xor(v, m, 32);
    if ((tid & 31) == 0) wsum[tid >> 5] = v;
    __syncthreads();
    if (tid < 8) {
        v = wsum[tid];
        #pragma unroll
        for (int m = 4; m >= 1; m >>= 1) v += __shfl_xor(v, m, 32);
        if (tid == 0) atomicAdd(lossAcc, v);
    }
    if (d == 0) atomicAdd(&counts[n], 1);
}

// new_cluster, total, smoothed, loss  (single block)
__global__ __launch_bounds__(1024) void k_cluster(const float* __restrict__ cs,
                                                  const int* __restrict__ counts,
                                                  const float* __restrict__ lossAcc,
                                                  float* __restrict__ outCluster,
                                                  float* __restrict__ outLoss,
                                                  float* __restrict__ smooth) {
    __shared__ float wsum[32];
    __shared__ float totalSh;
    int tid = threadIdx.x;
    float nc[8], s = 0.f;
    #pragma unroll
    for (int i = 0; i < 8; ++i) {
        int n = tid + i * 1024;
        nc[i] = DECAYF * cs[n] + (1.0f - DECAYF) * (float)counts[n];
        outCluster[n] = nc[i];
        s += nc[i];
    }
    #pragma unroll
    for (int m = 16; m >= 1; m >>= 1) s += __shfl_xor(s, m, 32);
    if ((tid & 31) == 0) wsum[tid >> 5] = s;
    __syncthreads();
    if (tid < 32) {
        s = wsum[tid];
        #pragma unroll
        for (int m = 16; m >= 1; m >>= 1) s += __shfl_xor(s, m, 32);
        if (tid == 0) totalSh = s;
    }
    __syncthreads();
    float total = totalSh;
    #pragma unroll
    for (int i = 0; i < 8; ++i) {
        int n = tid + i * 1024;
        smooth[n] = (nc[i] + EPSF) / (total + (float)NE * EPSF) * total;
    }
    if (tid == 0) outLoss[0] = lossAcc[0] * (1.0f / (float)NELEM);
}

__global__ __launch_bounds__(256) void k_embed(const float* __restrict__ eavg,
                                               const float* __restrict__ esum,
                                               const float* __restrict__ smooth,
                                               float* __restrict__ outEavg,
                                               float* __restrict__ outEmbed) {
    int e = blockIdx.x * 256 + threadIdx.x;
    if (e >= 64 * NE) return;
    float nea = DECAYF * eavg[e] + (1.0f - DECAYF) * esum[e];
    outEavg[e]  = nea;
    outEmbed[e] = nea / smooth[e & (NE - 1)];
}

extern "C" void kernel_launch(void* const* d_in, const int* in_sizes, int n_in,
                              void* d_out, int out_size, void* d_ws, size_t ws_size,
                              hipStream_t stream) {
    const float* input = (const float*)d_in[0];
    const float* embed = (const float*)d_in[1];
    const float* pos   = (const float*)d_in[2];
    const float* cs    = (const float*)d_in[3];
    const float* eavg  = (const float*)d_in[4];
    float* out = (float*)d_out;
    float* ws  = (float*)d_ws;

    int*   counts  = (int*)(ws + W_COUNTS);
    float* lossAcc = ws + W_LOSS;
    float* esum    = ws + W_ESUM;
    float* enorm   = ws + W_ENORM;
    int*   ind     = (int*)(ws + W_IND);
    float* smooth  = ws + W_SMOOTH;

    k_zero    <<<(ZERO_N + 255) / 256, 256, 0, stream>>>(ws, ZERO_N);
    k_enorm   <<<NE / 256,            256, 0, stream>>>(embed, enorm);
    k_prequant<<<NELEM / 256,         256, 0, stream>>>(input, pos, out + O_QUANT);
    k_argmax  <<<ROWS / (16 * 8),     256, 0, stream>>>(out + O_QUANT, embed, enorm,
                                                        ind, out + O_IND);
    k_scatter <<<ROWS / 4,            256, 0, stream>>>(input, embed, ind, out + O_QUANT,
                                                        esum, counts, lossAcc);
    k_cluster <<<1,                  1024, 0, stream>>>(cs, counts, lossAcc,
                                                        out + O_CLUSTER, out + O_LOSS, smooth);
    k_embed   <<<(64 * NE) / 256,     256, 0, stream>>>(eavg, esum, smooth,
                                                        out + O_EAVG, out + O_EMBED);
}